// SoluteDecoder_16630113370665
// MI455X (gfx1250) — compile-verified
//
#include <hip/hip_runtime.h>
#include <cstdint>

// ---------------------------------------------------------------------------
// Periodic-box k-NN top-32 (B=256, N=65536, D=3, k=32), one workgroup/batch.
// Single streaming pass (CDNA5 async global->LDS, 4-deep pipeline):
//   - per-particle 8-bit monotonic key (float bits of d^2 >> 24) stored in LDS
//   - lane-spread histogram (conflict-free) of the 64 exponent bins
// Then: threshold bin T (cum count crosses 32), collect candidates from the
// LDS key array (recompute exact d^2 from L2-hot gathers), 32 rounds of
// block argmin over (d2bits<<32|idx) keys (JAX-stable tie order), gather.
// ---------------------------------------------------------------------------

#define BLOCK    256
#define WAVES    8
#define TILE     128                 // particles per tile per wave
#define TILE_F   (TILE * 3)          // 384 floats = 1536 B
#define NBUF     4                   // async pipeline depth (tiles in flight)
#define NMAX     65536               // max particles per batch (LDS key array)
#define NBINS    64                  // clamped exponent bins (d^2 < 2 for unit box)
#define CAND_MAX 2048
#define KSEL     32

struct __align__(16) SMem {
    float              stage[WAVES][NBUF][TILE_F];   // 48 KB async staging
    unsigned           key32[NMAX / 4];              // 64 KB packed 8-bit keys
    unsigned           hist[NBINS * 32];             // 8 KB lane-spread histogram
    unsigned long long candKey[CAND_MAX];            // 16 KB (d2bits<<32)|idx
    unsigned long long red[BLOCK];                   // 2 KB scan/reduce scratch
    unsigned           winIdx[KSEL];
    unsigned           candCount;
    unsigned           selT;
};

// CDNA5 async copy: 16B global -> LDS per lane, tracked by ASYNCcnt (GVS mode).
__device__ __forceinline__ void async_ld16(unsigned lds_addr, unsigned goff, const void* sbase) {
    asm volatile("global_load_async_to_lds_b128 %0, %1, %2"
                 :
                 : "v"(lds_addr), "v"(goff), "s"(sbase)
                 : "memory");
}
#define WAIT_ASYNC_LE(n) asm volatile("s_wait_asynccnt " #n ::: "memory")

// Exact (non-contracted) minimum-image wrap; UNIT path: divide/multiply by 1.0
// are exact, so skipping them is bit-identical to the JAX elementwise chain.
template <bool UNIT>
__device__ __forceinline__ float wrap1(float c, float r, float bl) {
    float d = __fsub_rn(c, r);
    float q = UNIT ? d : __fdiv_rn(d, bl);
    float n = rintf(q);                       // round-half-even == jnp.round
    float m = UNIT ? n : __fmul_rn(bl, n);
    return __fsub_rn(d, m);
}

__device__ __forceinline__ float dist2(float x, float y, float z) {
    float xx = __fmul_rn(x, x);
    float yy = __fmul_rn(y, y);
    float zz = __fmul_rn(z, z);
    return __fadd_rn(__fadd_rn(xx, yy), zz);
}

template <bool UNIT>
__device__ void run_block(SMem& sm, const float* __restrict__ cb,
                          float rx, float ry, float rz,
                          float bx, float by, float bz,
                          float* __restrict__ out_kc, float* __restrict__ out_ki,
                          float* __restrict__ out_kd, int b, int N) {
    const int tid  = threadIdx.x;
    const int wave = tid >> 5;
    const int lane = tid & 31;

    // ---------------- init ----------------
    for (int i = tid; i < NBINS * 32; i += BLOCK) sm.hist[i] = 0u;
    if (tid == 0) sm.candCount = 0u;
    __syncthreads();

    // ---------------- single streaming pass (async -> LDS, 4-deep) --------
    const int perWave = N / WAVES;
    const int tiles   = perWave / TILE;
    const unsigned pstart = (unsigned)(wave * perWave);
    const void* gbase = (const void*)cb;
    const unsigned ldsBase = (unsigned)(size_t)(&sm.stage[wave][0][0]);

    auto issue = [&](int t) {
        int buf = t & (NBUF - 1);
        unsigned goff = (pstart + (unsigned)t * TILE) * 12u + (unsigned)lane * 16u;
        unsigned loff = ldsBase + (unsigned)buf * (TILE_F * 4u) + (unsigned)lane * 16u;
        async_ld16(loff,         goff,         gbase);
        async_ld16(loff + 512u,  goff + 512u,  gbase);
        async_ld16(loff + 1024u, goff + 1024u, gbase);
    };

    issue(0); issue(1); issue(2);
    for (int t = 0; t < tiles; ++t) {
        if (t + 3 < tiles) issue(t + 3);
        int ahead = tiles - 1 - t; if (ahead > 3) ahead = 3;
        if      (ahead == 3) WAIT_ASYNC_LE(9);
        else if (ahead == 2) WAIT_ASYNC_LE(6);
        else if (ahead == 1) WAIT_ASYNC_LE(3);
        else                 WAIT_ASYNC_LE(0);

        const float* st = sm.stage[wave][t & (NBUF - 1)];
        const unsigned pbase = pstart + (unsigned)t * TILE;
        unsigned packed = 0u;
#pragma unroll
        for (int q = 0; q < 4; ++q) {
            int pl = lane * 4 + q;                          // 4 consecutive particles/lane
            float x = wrap1<UNIT>(st[pl * 3 + 0], rx, bx);
            float y = wrap1<UNIT>(st[pl * 3 + 1], ry, by);
            float z = wrap1<UNIT>(st[pl * 3 + 2], rz, bz);
            float d2 = dist2(x, y, z);
            unsigned key = __float_as_uint(d2) >> 24;       // monotonic 8-bit prefix
            if (key > NBINS - 1) key = NBINS - 1;           // clamp (d^2<2 for unit box)
            atomicAdd(&sm.hist[key * 32 + lane], 1u);       // lane-spread: conflict-free
            packed |= key << (8 * q);
        }
        sm.key32[(pbase >> 2) + (unsigned)lane] = packed;   // coalesced packed store
    }
    __syncthreads();

    // ---------------- threshold bin T: inclusive cum crosses KSEL ---------
    unsigned cnt = 0u;
    if (tid < NBINS) {
#pragma unroll
        for (int j = 0; j < 32; ++j) cnt += sm.hist[tid * 32 + j];
    }
    sm.red[tid] = cnt;
    __syncthreads();
    for (int off = 1; off < NBINS; off <<= 1) {
        unsigned long long v  = sm.red[tid];
        unsigned long long v2 = (tid >= off && tid < NBINS) ? sm.red[tid - off] : 0ull;
        __syncthreads();
        if (tid < NBINS) sm.red[tid] = v + v2;
        __syncthreads();
    }
    if (tid < NBINS) {
        unsigned excl = (unsigned)sm.red[tid] - cnt;
        if (excl < KSEL && excl + cnt >= KSEL) sm.selT = (unsigned)tid;  // unique writer
    }
    __syncthreads();
    const unsigned T = sm.selT;

    // ---------------- collect candidates from LDS key array ---------------
    const unsigned words = (unsigned)N / 4u;
    for (unsigned w = tid; w < words; w += BLOCK) {
        unsigned v = sm.key32[w];
#pragma unroll
        for (int byt = 0; byt < 4; ++byt) {
            if (((v >> (8 * byt)) & 0xffu) <= T) {
                unsigned pos = atomicAdd(&sm.candCount, 1u);
                if (pos < CAND_MAX) {
                    unsigned idx = w * 4u + (unsigned)byt;
                    float x = wrap1<UNIT>(cb[(size_t)idx * 3 + 0], rx, bx);
                    float y = wrap1<UNIT>(cb[(size_t)idx * 3 + 1], ry, by);
                    float z = wrap1<UNIT>(cb[(size_t)idx * 3 + 2], rz, bz);
                    float d2 = dist2(x, y, z);
                    sm.candKey[pos] =
                        ((unsigned long long)__float_as_uint(d2) << 32) | (unsigned long long)idx;
                }
            }
        }
    }
    __syncthreads();

    unsigned M = sm.candCount;
    if (M > CAND_MAX) M = CAND_MAX;

    // ---------------- 32 rounds of block argmin (stable tie by index) -----
    for (int r = 0; r < KSEL; ++r) {
        unsigned long long best = ~0ull;
        for (unsigned i = tid; i < M; i += BLOCK) {
            unsigned long long k = sm.candKey[i];
            if (k < best) best = k;
        }
        sm.red[tid] = best;
        __syncthreads();
        for (int off = BLOCK / 2; off > 0; off >>= 1) {
            if (tid < off) {
                unsigned long long o = sm.red[tid + off];
                if (o < sm.red[tid]) sm.red[tid] = o;
            }
            __syncthreads();
        }
        const unsigned long long win = sm.red[0];
        // invalidate winner (keys are unique: idx embedded)
        for (unsigned i = tid; i < M; i += BLOCK) {
            if (sm.candKey[i] == win) sm.candKey[i] = ~0ull;
        }
        if (tid == 0) {
            unsigned pidx = (unsigned)(win & 0xffffffffull);
            unsigned db   = (unsigned)(win >> 32);
            out_kd[(size_t)b * KSEL + r] = __uint_as_float(db);
            out_ki[(size_t)b * KSEL + r] = (float)pidx;
            sm.winIdx[r] = pidx;
        }
        __syncthreads();
    }

    // ---------------- gather wrapped coords of winners ---------------------
    if (tid < KSEL) {
        unsigned pidx = sm.winIdx[tid];
        float x = wrap1<UNIT>(cb[(size_t)pidx * 3 + 0], rx, bx);
        float y = wrap1<UNIT>(cb[(size_t)pidx * 3 + 1], ry, by);
        float z = wrap1<UNIT>(cb[(size_t)pidx * 3 + 2], rz, bz);
        size_t o = ((size_t)b * KSEL + (size_t)tid) * 3;
        out_kc[o + 0] = x;
        out_kc[o + 1] = y;
        out_kc[o + 2] = z;
    }
}

__global__ __launch_bounds__(BLOCK)
void knn_topk_kernel(const float* __restrict__ ref,
                     const float* __restrict__ coords,
                     const float* __restrict__ boxL,
                     float* __restrict__ out,
                     int B, int N) {
    __shared__ SMem sm;
    const int b = blockIdx.x;

    const float rx = ref[b * 3 + 0], ry = ref[b * 3 + 1], rz = ref[b * 3 + 2];
    const float bx = boxL[0], by = boxL[1], bz = boxL[2];
    const float* cb = coords + (size_t)b * (size_t)N * 3;

    float* out_kc = out;
    float* out_ki = out + (size_t)B * KSEL * 3;
    float* out_kd = out + (size_t)B * KSEL * 3 + (size_t)B * KSEL;

    const bool unit = (bx == 1.0f) && (by == 1.0f) && (bz == 1.0f);
    if (unit) run_block<true >(sm, cb, rx, ry, rz, bx, by, bz, out_kc, out_ki, out_kd, b, N);
    else      run_block<false>(sm, cb, rx, ry, rz, bx, by, bz, out_kc, out_ki, out_kd, b, N);
}

extern "C" void kernel_launch(void* const* d_in, const int* in_sizes, int n_in,
                              void* d_out, int out_size, void* d_ws, size_t ws_size,
                              hipStream_t stream) {
    (void)n_in; (void)out_size; (void)d_ws; (void)ws_size;
    const float* ref    = (const float*)d_in[0];   // (B,1,3)
    const float* coords = (const float*)d_in[1];   // (B,N,3)
    const float* boxL   = (const float*)d_in[2];   // (3,)
    float* out = (float*)d_out;                    // [B*32*3 | B*32 | B*32] floats

    const int B = in_sizes[0] / 3;
    const int N = in_sizes[1] / (3 * B);
    if (N > NMAX || (N % (WAVES * TILE)) != 0) return;   // layout assumption guard

    knn_topk_kernel<<<dim3(B), dim3(BLOCK), 0, stream>>>(ref, coords, boxL, out, B, N);
}